// ts_cov_65455301591101
// MI455X (gfx1250) — compile-verified
//
#include <hip/hip_runtime.h>

typedef __attribute__((ext_vector_type(2))) float v2f;
typedef __attribute__((ext_vector_type(8))) float v8f;

#define B_ 256
#define T_ 512
#define F_ 32
#define W_ 5
#define S_ 508   // window starts: 0..507 (tail window already included)
#define C_ 496   // 32*31/2 pair covariances

// One wave32 = one (b, s) window. The 32x32 Gram matrix G = Xw^T * Xw
// (Xw = 5x32 window slice) is computed as three 16x16 WMMA tiles
// (0,0), (0,1), (1,1); K=5 is padded to two chained K=4 f32 WMMAs.
// cov[i,j] = G[i,j]/5 - (sx[i]/5)*(sx[j]/5), written with the
// triu_indices(32, k=1) linearization, non-temporal (pure output stream).
__global__ __launch_bounds__(256) void ts_cov_wmma_kernel(
    const float* __restrict__ x, float* __restrict__ out) {
  const int lane = threadIdx.x & 31;
  const int wid  = (int)((blockIdx.x * blockDim.x + threadIdx.x) >> 5);
  const int b = wid / S_;
  const int s = wid - b * S_;

  // Base of window row s in x[b, :, :]  (row stride = F_ floats)
  const float* __restrict__ xb = x + (size_t)b * T_ * F_ + (size_t)s * F_;

  // Per-feature window sum: lane L owns feature L.
  float sx = 0.f;
#pragma unroll
  for (int t = 0; t < W_; ++t) sx += xb[t * F_ + lane];

  // Hint: next window's fresh row (wave-uniform branch keeps EXEC intact).
  if (s + W_ < T_) __builtin_prefetch(xb + W_ * F_ + lane, 0, 1);

  const int half = lane >> 4;   // lane half selects K pair in A/B operands
  const int l15  = lane & 15;
  const int kb   = half * 2;    // first chunk: K = {0,1} | {2,3} per half

  float* __restrict__ ob = out + (size_t)wid * C_;

  const int TI[3] = {0, 0, 1};
  const int TJ[3] = {0, 1, 1};
#pragma unroll
  for (int t3 = 0; t3 < 3; ++t3) {
    const int I = TI[t3], J = TJ[t3];

    // A operand (16x4 f32): A[m,k] = x[s+k, 16I+m]; lane=m, VGPR/half=k.
    v2f a0, a1, b0, b1;
    a0.x = xb[kb * F_ + I * 16 + l15];
    a0.y = xb[(kb + 1) * F_ + I * 16 + l15];
    // B operand (4x16 f32): B[k,n] = x[s+k, 16J+n]; same K labeling as A.
    b0.x = xb[kb * F_ + J * 16 + l15];
    b0.y = xb[(kb + 1) * F_ + J * 16 + l15];
    // Second chunk K = {4,5}|{6,7}: only K=4 (lanes 0-15, first slot) is real.
    a1.x = (half == 0) ? xb[4 * F_ + I * 16 + l15] : 0.f;
    a1.y = 0.f;
    b1.x = (half == 0) ? xb[4 * F_ + J * 16 + l15] : 0.f;
    b1.y = 0.f;

    v8f acc = {};
    acc = __builtin_amdgcn_wmma_f32_16x16x4_f32(false, a0, false, b0,
                                                (short)0, acc, false, false);
    acc = __builtin_amdgcn_wmma_f32_16x16x4_f32(false, a1, false, b1,
                                                (short)0, acc, false, false);

    // C/D tile layout: lane = n + 16*(m>=8), VGPR r -> m = r + 8*(lane>>4).
    const int gn = J * 16 + l15;
    const float sxn = __shfl(sx, gn, 32);
#pragma unroll
    for (int r = 0; r < 8; ++r) {
      const int gm = I * 16 + r + 8 * half;
      const float sxm = __shfl(sx, gm, 32);   // uniform-shape shuffle (all lanes)
      if (gm < gn) {
        const int c = gm * 31 - (gm * (gm - 1)) / 2 + (gn - gm - 1);
        const float v = acc[r] * 0.2f - sxm * sxn * 0.04f;  // G/5 - sx*sy/25
        __builtin_nontemporal_store(v, &ob[c]);
      }
    }
  }
}

extern "C" void kernel_launch(void* const* d_in, const int* in_sizes, int n_in,
                              void* d_out, int out_size, void* d_ws, size_t ws_size,
                              hipStream_t stream) {
  (void)in_sizes; (void)n_in; (void)out_size; (void)d_ws; (void)ws_size;
  const float* x = (const float*)d_in[0];
  float* out = (float*)d_out;

  const int totalWaves    = B_ * S_;  // 130048
  const int wavesPerBlock = 256 / 32; // 8 waves per block (wave32)
  dim3 grid(totalWaves / wavesPerBlock);  // 16256 blocks, exact (EXEC all-ones)
  ts_cov_wmma_kernel<<<grid, 256, 0, stream>>>(x, out);
}